// ConstrainedExpectedSlicedPlan_37048387895358
// MI455X (gfx1250) — compile-verified
//
#include <hip/hip_runtime.h>
#include <math.h>

namespace {
constexpr int kB = 16;    // batch
constexpr int kN = 2048;  // points == ref points
constexpr int kD = 128;   // feature dim
constexpr int kL = 16;    // projections
}

typedef __attribute__((ext_vector_type(2))) float v2f;
typedef __attribute__((ext_vector_type(8))) float v8f;

// ---------------------------------------------------------------------------
// 1) weight_norm: W[l,:] = theta[l,:] / ||theta[l,:]||   (kL x kD)
// ---------------------------------------------------------------------------
__global__ __launch_bounds__(128) void k_norm_w(const float* __restrict__ theta,
                                                float* __restrict__ W) {
  const int l = blockIdx.x;
  const int d = threadIdx.x;
  float v = theta[l * kD + d];
  __shared__ float red[128];
  red[d] = v * v;
  __syncthreads();
  for (int s = 64; s > 0; s >>= 1) {
    if (d < s) red[d] += red[d + s];
    __syncthreads();
  }
  W[l * kD + d] = v * (1.0f / sqrtf(red[0]));
}

// ---------------------------------------------------------------------------
// 2) Projection GEMM on the matrix pipe: S = [X ; ref] (34816 x 128) * W^T
//    (128 x 16), f32-exact via V_WMMA_F32_16X16X4_F32, one wave per 16-row
//    M-tile, K reduced in 32 steps of 4. Outputs stored column-major
//    (one contiguous 2048-vector per (b,l) / l column) for the sort kernels.
//    A layout (16x4 f32): lane<16 holds row m=lane, {K0,K1}; lane>=16 holds
//    row m=lane-16, {K2,K3}. B (4x16): mirrored K split, n = lane&15.
//    C/D (16x16 f32): vgpr v -> M = v + 8*(lane>>4), N = lane&15.
// ---------------------------------------------------------------------------
__global__ __launch_bounds__(32) void k_slices(const float* __restrict__ X,
                                               const float* __restrict__ ref,
                                               const float* __restrict__ W,
                                               float* __restrict__ Xsl,
                                               float* __restrict__ Rsl) {
  const int lane  = threadIdx.x;
  const int m     = lane & 15;   // row within tile (A) / column l (B, C)
  const int khalf = lane >> 4;   // K-half select
  const int tile  = blockIdx.x;
  const int xTiles = (kB * kN) / 16;   // 2048 tiles of X rows, then ref tiles

  const float* abase;
  int r0;
  if (tile < xTiles) { abase = X;   r0 = tile * 16; }
  else               { abase = ref; r0 = (tile - xTiles) * 16; }

  const float* arow = abase + (size_t)(r0 + m) * kD;
  const float* wrow = W + (size_t)m * kD;   // B column n = lane&15

  v8f c = {};
#pragma unroll 4
  for (int k0 = 0; k0 < kD; k0 += 4) {
    v2f a, b;
    a.x = arow[k0 + 2 * khalf + 0];
    a.y = arow[k0 + 2 * khalf + 1];
    b.x = wrow[k0 + 2 * khalf + 0];
    b.y = wrow[k0 + 2 * khalf + 1];
    c = __builtin_amdgcn_wmma_f32_16x16x4_f32(false, a, false, b,
                                              (short)0, c, false, false);
  }

  if (tile < xTiles) {
#pragma unroll
    for (int v = 0; v < 8; ++v) {
      int grow = tile * 16 + v + 8 * khalf;   // global X row
      int b_ = grow >> 11;                    // / kN
      int nn = grow & (kN - 1);
      Xsl[((size_t)(b_ * kL + m)) * kN + nn] = c[v];
    }
  } else {
#pragma unroll
    for (int v = 0; v < 8; ++v) {
      int rr = r0 + v + 8 * khalf;
      Rsl[(size_t)m * kN + rr] = c[v];
    }
  }
}

// ---------------------------------------------------------------------------
// 3) Bitonic argsort of one 2048-float column in LDS (16 KB of the WGP's
//    320 KB). Blocks 0..B*L-1 sort Xsl columns -> Xind; the last L blocks
//    sort Rsl columns -> Rind and the inverse permutation rankR.
// ---------------------------------------------------------------------------
__global__ __launch_bounds__(256) void k_sort(const float* __restrict__ Xsl,
                                              const float* __restrict__ Rsl,
                                              int* __restrict__ Xind,
                                              int* __restrict__ Rind,
                                              int* __restrict__ rankR) {
  __shared__ float sv[kN];
  __shared__ int   si[kN];
  const int c = blockIdx.x;
  const float* col = (c < kB * kL) ? (Xsl + (size_t)c * kN)
                                   : (Rsl + (size_t)(c - kB * kL) * kN);
  for (int i = threadIdx.x; i < kN; i += 256) { sv[i] = col[i]; si[i] = i; }
  __syncthreads();

  for (int k = 2; k <= kN; k <<= 1) {
    for (int j = k >> 1; j > 0; j >>= 1) {
      for (int t = threadIdx.x; t < kN; t += 256) {
        int p = t ^ j;
        if (p > t) {
          bool up = ((t & k) == 0);
          float a = sv[t], b = sv[p];
          if (up ? (a > b) : (a < b)) {
            sv[t] = b; sv[p] = a;
            int ti = si[t]; si[t] = si[p]; si[p] = ti;
          }
        }
      }
      __syncthreads();
    }
  }

  if (c < kB * kL) {
    int* o = Xind + (size_t)c * kN;
    for (int i = threadIdx.x; i < kN; i += 256) o[i] = si[i];
  } else {
    const int l = c - kB * kL;
    for (int i = threadIdx.x; i < kN; i += 256) {
      Rind[(size_t)l * kN + i]      = si[i];
      rankR[(size_t)l * kN + si[i]] = i;   // inverse permutation
    }
  }
}

// ---------------------------------------------------------------------------
// 4) exact_dist[b,l] = (1/N) * sum_i || ref[Rind[i,l]] - X[b, Xind[b,i,l]] ||
//    One block per (b,l); rank-paired gathers, float4 vector loads (all of X
//    and ref sit in the 192 MB L2).
// ---------------------------------------------------------------------------
__global__ __launch_bounds__(256) void k_dist(const float* __restrict__ X,
                                              const float* __restrict__ ref,
                                              const int* __restrict__ Xind,
                                              const int* __restrict__ Rind,
                                              float* __restrict__ dist) {
  const int bl = blockIdx.x;          // b*kL + l
  const int b = bl / kL, l = bl % kL;
  const int* xi = Xind + (size_t)bl * kN;
  const int* ri = Rind + (size_t)l * kN;

  float acc = 0.f;
  for (int i = threadIdx.x; i < kN; i += 256) {
    const float4* xr = (const float4*)(X + ((size_t)b * kN + xi[i]) * kD);
    const float4* rr = (const float4*)(ref + (size_t)ri[i] * kD);
    float s = 0.f;
#pragma unroll 8
    for (int q = 0; q < kD / 4; ++q) {
      float4 a = xr[q], c = rr[q];
      float d0 = a.x - c.x, d1 = a.y - c.y, d2 = a.z - c.z, d3 = a.w - c.w;
      s = fmaf(d0, d0, fmaf(d1, d1, fmaf(d2, d2, fmaf(d3, d3, s))));
    }
    acc += sqrtf(s);
  }

  __shared__ float red[256];
  red[threadIdx.x] = acc;
  __syncthreads();
  for (int s2 = 128; s2 > 0; s2 >>= 1) {
    if (threadIdx.x < s2) red[threadIdx.x] += red[threadIdx.x + s2];
    __syncthreads();
  }
  if (threadIdx.x == 0) dist[bl] = red[0] / (float)kN;
}

// ---------------------------------------------------------------------------
// 5) weights[b,:] = softmax(-dist[b,:]); also zero the second (L,) output.
// ---------------------------------------------------------------------------
__global__ __launch_bounds__(32) void k_softmax(const float* __restrict__ dist,
                                                float* __restrict__ wts,
                                                float* __restrict__ tail) {
  const int b = blockIdx.x;
  const int t = threadIdx.x;
  if (b == 0 && t < kL) tail[t] = 0.0f;
  if (t == 0) {
    float v[kL];
    float m = -INFINITY;
    for (int l = 0; l < kL; ++l) { v[l] = -dist[b * kL + l]; m = fmaxf(m, v[l]); }
    float s = 0.f;
    for (int l = 0; l < kL; ++l) { v[l] = expf(v[l] - m); s += v[l]; }
    float inv = 1.0f / s;
    for (int l = 0; l < kL; ++l) wts[b * kL + l] = v[l] * inv;
  }
}

// ---------------------------------------------------------------------------
// 6) embeddings[b,r,:] = scale * sum_l w[b,l]*X[b, Xind[b, rankR[l][r], l], :]
//                        - ref[r,:]
//    (scale = (1/N)/(1/N + 1e-8): the plan rows each sum to exactly 1/N.)
//    One block per (b,r); 16 gather indices staged through LDS.
// ---------------------------------------------------------------------------
__global__ __launch_bounds__(128) void k_embed(const float* __restrict__ X,
                                               const float* __restrict__ ref,
                                               const int* __restrict__ Xind,
                                               const int* __restrict__ rankR,
                                               const float* __restrict__ wts,
                                               float* __restrict__ out) {
  const int br = blockIdx.x;
  const int b = br >> 11;            // / kN
  const int r = br & (kN - 1);
  const int d = threadIdx.x;

  __shared__ float wloc[kL];
  __shared__ int   xloc[kL];
  if (d < kL) {
    wloc[d] = wts[b * kL + d];
    int i   = rankR[(size_t)d * kN + r];
    xloc[d] = Xind[((size_t)(b * kL + d)) * kN + i];
  }
  __syncthreads();

  float acc = 0.f;
#pragma unroll
  for (int l = 0; l < kL; ++l)
    acc = fmaf(wloc[l], X[((size_t)b * kN + xloc[l]) * kD + d], acc);

  constexpr float scale = 1.0f / (1.0f + (float)kN * 1e-8f);
  out[((size_t)b * kN + r) * kD + d] = acc * scale - ref[(size_t)r * kD + d];
}

// ---------------------------------------------------------------------------
extern "C" void kernel_launch(void* const* d_in, const int* in_sizes, int n_in,
                              void* d_out, int out_size, void* d_ws, size_t ws_size,
                              hipStream_t stream) {
  (void)in_sizes; (void)n_in; (void)out_size; (void)ws_size;

  const float* X     = (const float*)d_in[0];  // (B,N,D)
  const float* ref   = (const float*)d_in[1];  // (N,D)
  const float* theta = (const float*)d_in[2];  // (L,D)
  float* out = (float*)d_out;                  // (B,N,D) ++ (L,)

  char* ws = (char*)d_ws;
  size_t off = 0;
  auto take = [&](size_t bytes) -> char* {
    char* p = ws + off;
    off = (off + bytes + 255) & ~(size_t)255;
    return p;
  };
  float* W     = (float*)take((size_t)kL * kD * sizeof(float));        //   8 KB
  float* Xsl   = (float*)take((size_t)kB * kL * kN * sizeof(float));   //   2 MB
  float* Rsl   = (float*)take((size_t)kL * kN * sizeof(float));        // 128 KB
  int*   Xind  = (int*)  take((size_t)kB * kL * kN * sizeof(int));     //   2 MB
  int*   Rind  = (int*)  take((size_t)kL * kN * sizeof(int));          // 128 KB
  int*   rankR = (int*)  take((size_t)kL * kN * sizeof(int));          // 128 KB
  float* dist  = (float*)take((size_t)kB * kL * sizeof(float));
  float* wts   = (float*)take((size_t)kB * kL * sizeof(float));

  k_norm_w<<<kL, 128, 0, stream>>>(theta, W);

  const int xTiles = (kB * kN) / 16;
  const int rTiles = kN / 16;
  k_slices<<<xTiles + rTiles, 32, 0, stream>>>(X, ref, W, Xsl, Rsl);

  k_sort<<<kB * kL + kL, 256, 0, stream>>>(Xsl, Rsl, Xind, Rind, rankR);

  k_dist<<<kB * kL, 256, 0, stream>>>(X, ref, Xind, Rind, dist);

  k_softmax<<<kB, 32, 0, stream>>>(dist, wts, out + (size_t)kB * kN * kD);

  k_embed<<<kB * kN, 128, 0, stream>>>(X, ref, Xind, rankR, wts, out);
}